// RPNLoss_28278064677322
// MI455X (gfx1250) — compile-verified
//
#include <hip/hip_runtime.h>

// ---------------- problem constants (from reference setup) ----------------
#define BATCH   8
#define NA      9            // anchors per cell
#define FH      64
#define FW      64
#define NG      64           // GT boxes per batch
#define NPLANE  (FH*FW)      // 4096
#define NTOT    (BATCH*NA*NPLANE)   // 294912
#define NBLK    (NTOT/256)          // 1152
#define HBINS   256
#define EPSF    1e-6f
#define NCLS_SCALE   2048.0f        // B * TOTAL_ANCHORS
#define NREG_SCALE   368640.0f      // LAMBDA * fh*fw*A = 10 * 36864

// ---------------- CDNA5 address-space helpers ----------------
typedef __attribute__((address_space(1))) int gint;   // global
typedef __attribute__((address_space(3))) int lint;   // LDS

static __device__ __forceinline__ gint* as_global(const void* p) {
    // flat address of global memory == global address on AMDGPU
    return (gint*)(unsigned long long)p;
}
static __device__ __forceinline__ lint* as_lds(void* p) {
    // flat LDS address = {SHARED_BASE, offset[31:0]} -> low 32 bits are the LDS offset
    return (lint*)(unsigned)(unsigned long long)p;
}

typedef float v2f __attribute__((ext_vector_type(2)));
typedef float v8f __attribute__((ext_vector_type(8)));

// anchor w/h table: scales {128,256,512} x ratios {0.5,1,2}; w=s*sqrt(r), h=s/sqrt(r)
__device__ __constant__ float c_aw[NA] = {90.50966799f,128.0f,181.01933598f,
                                          181.01933598f,256.0f,362.03867197f,
                                          362.03867197f,512.0f,724.07734394f};
__device__ __constant__ float c_ah[NA] = {181.01933598f,128.0f,90.50966799f,
                                          362.03867197f,256.0f,181.01933598f,
                                          724.07734394f,512.0f,362.03867197f};

static __device__ __forceinline__ float hash01(unsigned v, unsigned salt) {
    unsigned x = v * 0x9E3779B1u + salt;
    x ^= x >> 16; x *= 0x7FEB352Du;
    x ^= x >> 15; x *= 0x846CA68Bu;
    x ^= x >> 16;
    return (float)(x >> 8) * (1.0f / 16777216.0f);   // [0,1)
}
static __device__ __forceinline__ int bin_of(float p) {
    int b = (int)((p - 0.1f) * (HBINS / 0.9f));
    return b < 0 ? 0 : (b > HBINS - 1 ? HBINS - 1 : b);
}
static __device__ __forceinline__ float sl1(float d) {
    float ad = fabsf(d);
    return ad < 1.0f ? 0.5f * ad * ad : ad - 0.5f;
}

// ---------------- K0: zero accumulator / thresholds / histograms ----------------
__global__ void k_init(unsigned* __restrict__ ws) {
    ws[blockIdx.x * 256 + threadIdx.x] = 0u;   // first 1024 u32 of workspace
}

// ---------------- K1: IoU match + pos/neg + priority histograms ----------------
__global__ __launch_bounds__(256) void k_match(
    const float* __restrict__ boxes, const int* __restrict__ pimh,
    const int* __restrict__ pimw,
    unsigned* __restrict__ meta, unsigned* __restrict__ histP,
    unsigned* __restrict__ histN)
{
    __shared__ float    sBox[NG * 4];
    __shared__ unsigned sHP[HBINS];
    __shared__ unsigned sHN[HBINS];

    const int tid   = threadIdx.x;
    const int idx   = blockIdx.x * 256 + tid;
    const int plane = idx >> 12;               // one (b,a) plane per 4096 idx
    const int b     = plane / NA;
    const int a     = plane - b * NA;
    const int pix   = idx & (NPLANE - 1);
    const int y     = pix >> 6;
    const int x     = pix & 63;

    sHP[tid] = 0u;
    sHN[tid] = 0u;

    // stage this batch's 64 boxes (256 floats) into LDS with the async DMA path
    const float* gsrc = boxes + (size_t)b * NG * 4 + tid;
    __builtin_amdgcn_global_load_async_to_lds_b32(as_global(gsrc), as_lds(&sBox[tid]), 0, 0);

    // overlap: anchor geometry while the async copy is in flight
    const float imh = (float)*pimh, imw = (float)*pimw;
    const float acx = (x + 0.5f) * (imw * (1.0f / FW));
    const float acy = (y + 0.5f) * (imh * (1.0f / FH));
    const float aw  = c_aw[a], ah = c_ah[a];
    const float ax1 = acx - 0.5f * aw, ay1 = acy - 0.5f * ah;
    const float ax2 = acx + 0.5f * aw, ay2 = acy + 0.5f * ah;
    const bool  inside = (ax1 >= 0.0f) && (ay1 >= 0.0f) && (ax2 <= imw) && (ay2 <= imh);
    const float areaA  = aw * ah;

    asm volatile("s_wait_asynccnt 0" ::: "memory");
    __syncthreads();

    float best = -1.0f; int bidx = 0;
    #pragma unroll 8
    for (int g = 0; g < NG; ++g) {
        const float bcx = sBox[g * 4 + 0], bcy = sBox[g * 4 + 1];
        const float bw  = sBox[g * 4 + 2], bh  = sBox[g * 4 + 3];
        const float bx1 = bcx - 0.5f * bw, by1 = bcy - 0.5f * bh;
        const float bx2 = bcx + 0.5f * bw, by2 = bcy + 0.5f * bh;
        const float iw  = fminf(ax2, bx2) - fmaxf(ax1, bx1);
        const float ih  = fminf(ay2, by2) - fmaxf(ay1, by1);
        const float inter = fmaxf(iw, 0.0f) * fmaxf(ih, 0.0f);
        const float iou = inter / (areaA + bw * bh - inter + EPSF);
        if (iou > best) { best = iou; bidx = g; }
    }

    const bool pos = inside && (best >= 0.7f);
    const bool neg = inside && (best < 0.3f);
    unsigned binp = 0u, binn = 0u, flags = 0u;
    if (pos) {
        binp = (unsigned)bin_of(0.1f + 0.9f * hash01((unsigned)idx, 0x9E3779B9u));
        flags |= 1u;
        atomicAdd(&sHP[binp], 1u);
    }
    if (neg) {
        binn = (unsigned)bin_of(0.1f + 0.9f * hash01((unsigned)idx, 0x85EBCA6Bu));
        flags |= 2u;
        atomicAdd(&sHN[binn], 1u);
    }
    meta[idx] = binp | (binn << 8) | ((unsigned)bidx << 16) | (flags << 24);

    __syncthreads();
    if (sHP[tid]) atomicAdd(&histP[tid], sHP[tid]);
    if (sHN[tid]) atomicAdd(&histN[tid], sHN[tid]);
}

// ---------------- K2: radix-select thresholds from histograms ----------------
__global__ void k_select(const unsigned* __restrict__ histP,
                         const unsigned* __restrict__ histN,
                         int* __restrict__ thr)
{
    if (threadIdx.x != 0) return;
    int totalP = 0, totalN = 0;
    for (int i = 0; i < HBINS; ++i) { totalP += (int)histP[i]; totalN += (int)histN[i]; }

    const int kp = totalP < 128 ? totalP : 128;          // n_pos (matches min(#pos,128))
    int tp = HBINS, run = 0;
    if (kp > 0) {
        for (int t = HBINS - 1; t >= 0; --t) { run += (int)histP[t]; if (run >= kp) { tp = t; break; } }
    }
    int kn = 256 - kp;
    if (kn > totalN) kn = totalN;
    int tn = HBINS; run = 0;
    if (kn > 0) {
        for (int t = HBINS - 1; t >= 0; --t) { run += (int)histN[t]; if (run >= kn) { tn = t; break; } }
    }
    thr[0] = tp;
    thr[1] = tn;
}

// ---------------- K3: CE + smooth-L1 loss, WMMA wave reduction ----------------
__global__ __launch_bounds__(256) void k_loss(
    const float* __restrict__ in, const float* __restrict__ boxes,
    const int* __restrict__ pimh, const int* __restrict__ pimw,
    const unsigned* __restrict__ meta, const int* __restrict__ thr,
    float* __restrict__ acc)
{
    __shared__ float sBox[NG * 4];
    __shared__ float sWave[8];

    const int tid   = threadIdx.x;
    const int idx   = blockIdx.x * 256 + tid;
    const int plane = idx >> 12;
    const int b     = plane / NA;
    const int a     = plane - b * NA;
    const int pix   = idx & (NPLANE - 1);
    const int y     = pix >> 6;
    const int x     = pix & 63;

    const float* gsrc = boxes + (size_t)b * NG * 4 + tid;
    __builtin_amdgcn_global_load_async_to_lds_b32(as_global(gsrc), as_lds(&sBox[tid]), 0, 0);

    // selection flags from packed metadata (overlaps async copy)
    const int tp = thr[0], tn = thr[1];
    const unsigned m   = meta[idx];
    const int binp     = (int)(m & 255u);
    const int binn     = (int)((m >> 8) & 255u);
    const int gi       = (int)((m >> 16) & 255u);
    const unsigned fl  = m >> 24;
    const bool selp = (fl & 1u) && (binp >= tp);
    const bool seln = (fl & 2u) && (binn >= tn);

    // classification CE: 2-way log_softmax over channels {a*6+0, a*6+1}
    const size_t base = (size_t)(b * (NA * 6) + a * 6) * NPLANE + pix;
    const float l0 = in[base];
    const float l1 = in[base + NPLANE];
    const float mx = fmaxf(l0, l1);
    const float lse = mx + __logf(__expf(l0 - mx) + __expf(l1 - mx));
    const float ce  = lse - (selp ? l1 : l0);
    const float cls_term = (selp || seln) ? ce : 0.0f;

    asm volatile("s_wait_asynccnt 0" ::: "memory");
    __syncthreads();

    // regression smooth-L1 on (t - t*) for the matched GT box
    const float imh = (float)*pimh, imw = (float)*pimw;
    const float aw  = c_aw[a], ah = c_ah[a];
    (void)imh; (void)imw; (void)x; (void)y;
    const float pcx = in[base + 2 * NPLANE];
    const float pcy = in[base + 3 * NPLANE];
    const float pw  = in[base + 4 * NPLANE];
    const float ph  = in[base + 5 * NPLANE];
    const float gcx = sBox[gi * 4 + 0], gcy = sBox[gi * 4 + 1];
    const float gw  = sBox[gi * 4 + 2], gh  = sBox[gi * 4 + 3];
    const float dx = (pcx - gcx) / aw;                                   // tx - tx*
    const float dy = (pcy - gcy) / ah;                                   // ty - ty*
    const float dw = __logf(fmaxf(pw, EPSF)) - __logf(fmaxf(gw, EPSF));  // tw - tw*
    const float dh = __logf(fmaxf(ph, EPSF)) - __logf(fmaxf(gh, EPSF));  // th - th*
    const float r  = sl1(dx) + sl1(dy) + sl1(dw) + sl1(dh);
    const float reg_term = selp ? r : 0.0f;

    const float contrib = NCLS_SCALE * cls_term + NREG_SCALE * reg_term;

    // ---- wave32 sum via V_WMMA_F32_16X16X4_F32: A = per-lane contribs, B = ones
    // A[m][0]=v_m, A[m][2]=v_{m+16}, rest 0  =>  D[m][n] = v_m + v_{m+16}
    v2f af; af.x = contrib; af.y = 0.0f;
    v2f bf; bf.x = 1.0f;    bf.y = 1.0f;
    v8f cf = {};
    v8f d8 = __builtin_amdgcn_wmma_f32_16x16x4_f32(false, af, false, bf,
                                                   (short)0, cf, false, false);
    float s = d8[0] + d8[1] + d8[2] + d8[3] + d8[4] + d8[5] + d8[6] + d8[7];
    s += __shfl_xor(s, 16);        // lanes 0-15 hold M=0..7 sums, 16-31 hold M=8..15

    const int lane = tid & 31, wv = tid >> 5;
    if (lane == 0) sWave[wv] = s;
    __syncthreads();
    if (tid == 0) {
        float t = 0.0f;
        #pragma unroll
        for (int i = 0; i < 8; ++i) t += sWave[i];
        atomicAdd(acc, t);
    }
}

// ---------------- K4: publish scalar ----------------
__global__ void k_final(const float* __restrict__ acc, float* __restrict__ out) {
    out[0] = acc[0];
}

// ---------------- host-side launcher ----------------
extern "C" void kernel_launch(void* const* d_in, const int* in_sizes, int n_in,
                              void* d_out, int out_size, void* d_ws, size_t ws_size,
                              hipStream_t stream) {
    (void)in_sizes; (void)n_in; (void)out_size; (void)ws_size;
    const float* input = (const float*)d_in[0];   // (8, 54, 64, 64) f32
    const float* boxes = (const float*)d_in[1];   // (8, 64, 4)      f32
    const int*   pimh  = (const int*)d_in[2];     // scalar 1024
    const int*   pimw  = (const int*)d_in[3];     // scalar 1024

    unsigned* wsu   = (unsigned*)d_ws;
    float*    acc   = (float*)d_ws;          // [0]
    int*      thr   = (int*)d_ws + 4;        // [4],[5]
    unsigned* histP = wsu + 16;              // 256 bins
    unsigned* histN = wsu + 272;             // 256 bins
    unsigned* meta  = wsu + 1024;            // NTOT packed words

    k_init  <<<4,    256, 0, stream>>>(wsu);
    k_match <<<NBLK, 256, 0, stream>>>(boxes, pimh, pimw, meta, histP, histN);
    k_select<<<1,    64,  0, stream>>>(histP, histN, thr);
    k_loss  <<<NBLK, 256, 0, stream>>>(input, boxes, pimh, pimw, meta, thr, acc);
    k_final <<<1,    1,   0, stream>>>(acc, (float*)d_out);
}